// GaussRenderer_40973988004321
// MI455X (gfx1250) — compile-verified
//
#include <hip/hip_runtime.h>
#include <math.h>

#define NG   8192
#define IMW  128
#define IMH  128
#define TIL  16

// ws layout (bytes)
#define PRE_OFF   0u                 // 8192 * 16 f32  (depth,m2,conic,op,rect,rgb)
#define SIDX_OFF  524288u            // 8192 i32
#define SG_OFF    557056u            // 8192 * 12 f32 (sorted render structs)
#define BP_OFF    950272u            // 2048 * 16 * 4 f32 (WMMA B panels)

typedef __attribute__((ext_vector_type(2))) float v2f;
typedef __attribute__((ext_vector_type(4))) float v4f;
typedef __attribute__((ext_vector_type(8))) float v8f;

__device__ __forceinline__ float clampf(float v, float lo, float hi) {
  return fminf(fmaxf(v, lo), hi);
}

// tan(pi/6), focal = 128 / (2*tan(pi/6))
#define TANF   0.5773502691896257f
#define FOCAL  110.85125168440814f

// ------------------------------------------------------------------
// Kernel 1: per-gaussian preprocess
// pre[n*16 + {0:depth,1:m2x,2:m2y,3:c00,4:c11,5:cxy,6:op,
//             7:rminx,8:rminy,9:rmaxx,10:rmaxy,11:r,12:g,13:b}]
// ------------------------------------------------------------------
__global__ __launch_bounds__(256) void gs_preprocess(
    const float* __restrict__ means3D, const float* __restrict__ opacity,
    const float* __restrict__ scales,  const float* __restrict__ rotations,
    const float* __restrict__ shs,     const float* __restrict__ vm,
    const float* __restrict__ pm,      const float* __restrict__ camc,
    float* __restrict__ pre) {
  const int n = blockIdx.x * blockDim.x + threadIdx.x;
  if (n >= NG) return;

  const float mx = means3D[n*3+0], my = means3D[n*3+1], mz = means3D[n*3+2];

  // row-vector convention: pv[j] = sum_i p[i]*V[i][j], p = (mx,my,mz,1)
  float pv[4], ph[4];
#pragma unroll
  for (int j = 0; j < 4; ++j)
    pv[j] = mx*vm[0*4+j] + my*vm[1*4+j] + mz*vm[2*4+j] + vm[3*4+j];
#pragma unroll
  for (int j = 0; j < 4; ++j)
    ph[j] = pv[0]*pm[0*4+j] + pv[1]*pm[1*4+j] + pv[2]*pm[2*4+j] + pv[3]*pm[3*4+j];

  const float invw = 1.0f / (ph[3] + 1e-6f);
  const float ppx = ph[0] * invw, ppy = ph[1] * invw;
  const float m2x = ((ppx + 1.0f) * (float)IMW - 1.0f) * 0.5f;
  const float m2y = ((ppy + 1.0f) * (float)IMH - 1.0f) * 0.5f;
  const float depth = pv[2];

  // ---- SH degree 3 (direction NOT normalized, matching reference) ----
  const float dx = mx - camc[0], dy = my - camc[1], dz = mz - camc[2];
  const float x = dx, y = dy, z = dz;
  const float xx = x*x, yy = y*y, zz = z*z;
  const float xy = x*y, yz = y*z, xz = x*z;
  float coef[16];
  coef[0]  = 0.28209479177387814f;
  coef[1]  = -0.4886025119029199f * y;
  coef[2]  =  0.4886025119029199f * z;
  coef[3]  = -0.4886025119029199f * x;
  coef[4]  =  1.0925484305920792f * xy;
  coef[5]  = -1.0925484305920792f * yz;
  coef[6]  =  0.31539156525252005f * (2.0f*zz - xx - yy);
  coef[7]  = -1.0925484305920792f * xz;
  coef[8]  =  0.5462742152960396f * (xx - yy);
  coef[9]  = -0.5900435899266435f * y * (3.0f*xx - yy);
  coef[10] =  2.890611442640554f  * xy * z;
  coef[11] = -0.4570457994644658f * y * (4.0f*zz - xx - yy);
  coef[12] =  0.3731763325901154f * z * (2.0f*zz - 3.0f*xx - 3.0f*yy);
  coef[13] = -0.4570457994644658f * x * (4.0f*zz - xx - yy);
  coef[14] =  1.445305721320277f  * z * (xx - yy);
  coef[15] = -0.5900435899266435f * x * (xx - 3.0f*yy);
  float col[3] = {0.f, 0.f, 0.f};
#pragma unroll
  for (int j = 0; j < 16; ++j) {
    const float cj = coef[j];
    col[0] += cj * shs[(n*16 + j)*3 + 0];
    col[1] += cj * shs[(n*16 + j)*3 + 1];
    col[2] += cj * shs[(n*16 + j)*3 + 2];
  }
#pragma unroll
  for (int c = 0; c < 3; ++c) col[c] = fmaxf(col[c] + 0.5f, 0.0f);

  // ---- quaternion -> R, L = R*diag(s), cov3d = L L^T ----
  float qr = rotations[n*4+0], qx = rotations[n*4+1];
  float qy = rotations[n*4+2], qz = rotations[n*4+3];
  const float qn = rsqrtf(qr*qr + qx*qx + qy*qy + qz*qz);
  qr *= qn; qx *= qn; qy *= qn; qz *= qn;
  float R[3][3];
  R[0][0] = 1.f - 2.f*(qy*qy + qz*qz); R[0][1] = 2.f*(qx*qy - qr*qz); R[0][2] = 2.f*(qx*qz + qr*qy);
  R[1][0] = 2.f*(qx*qy + qr*qz); R[1][1] = 1.f - 2.f*(qx*qx + qz*qz); R[1][2] = 2.f*(qy*qz - qr*qx);
  R[2][0] = 2.f*(qx*qz - qr*qy); R[2][1] = 2.f*(qy*qz + qr*qx); R[2][2] = 1.f - 2.f*(qx*qx + qy*qy);
  const float s0 = scales[n*3+0], s1 = scales[n*3+1], s2 = scales[n*3+2];
  float L[3][3];
#pragma unroll
  for (int i = 0; i < 3; ++i) { L[i][0] = R[i][0]*s0; L[i][1] = R[i][1]*s1; L[i][2] = R[i][2]*s2; }
  float C3[3][3];
#pragma unroll
  for (int i = 0; i < 3; ++i)
#pragma unroll
    for (int k = 0; k < 3; ++k)
      C3[i][k] = L[i][0]*L[k][0] + L[i][1]*L[k][1] + L[i][2]*L[k][2];

  // M = Wm C Wm^T with Wm = V3^T  => tmp[i][k] = sum_j V[j][i] C[j][k]; M[i][l] = sum_k tmp[i][k] V[k][l]
  float tmpM[3][3], M[3][3];
#pragma unroll
  for (int i = 0; i < 3; ++i)
#pragma unroll
    for (int k = 0; k < 3; ++k)
      tmpM[i][k] = vm[0*4+i]*C3[0][k] + vm[1*4+i]*C3[1][k] + vm[2*4+i]*C3[2][k];
#pragma unroll
  for (int i = 0; i < 3; ++i)
#pragma unroll
    for (int l = 0; l < 3; ++l)
      M[i][l] = tmpM[i][0]*vm[0*4+l] + tmpM[i][1]*vm[1*4+l] + tmpM[i][2]*vm[2*4+l];

  // ---- J (EWA) using clipped tx, ty ----
  const float tz = pv[2];
  const float lim = 1.3f * TANF;
  const float txc = clampf(pv[0] / tz, -lim, lim) * tz;
  const float tyc = clampf(pv[1] / tz, -lim, lim) * tz;
  const float j00 = FOCAL / tz;
  const float j11 = FOCAL / tz;
  const float j02 = -txc * FOCAL / (tz * tz);
  const float j12 = -tyc * FOCAL / (tz * tz);
  float u0[3], u1[3];
#pragma unroll
  for (int k = 0; k < 3; ++k) { u0[k] = j00*M[0][k] + j02*M[2][k]; u1[k] = j11*M[1][k] + j12*M[2][k]; }
  const float a  = u0[0]*j00 + u0[2]*j02 + 0.3f;
  const float b  = u0[1]*j11 + u0[2]*j12;
  const float cc = u1[0]*j00 + u1[2]*j02;
  const float d  = u1[1]*j11 + u1[2]*j12 + 0.3f;

  // radii / rect
  const float det = a*d - b*cc;
  const float mid = 0.5f * (a + d);
  const float sq  = sqrtf(fmaxf(mid*mid - det, 0.1f));
  const float lam = fmaxf(mid + sq, mid - sq);
  const float radii = 3.0f * ceilf(sqrtf(lam));
  const float rminx = clampf(m2x - radii, 0.f, (float)(IMW-1));
  const float rminy = clampf(m2y - radii, 0.f, (float)(IMH-1));
  const float rmaxx = clampf(m2x + radii, 0.f, (float)(IMW-1));
  const float rmaxy = clampf(m2y + radii, 0.f, (float)(IMH-1));

  // conic
  const float idet = 1.0f / det;
  const float c00 = d * idet;
  const float c11 = a * idet;
  const float cxy = -(b + cc) * idet;   // c01 + c10

  float* o = pre + (size_t)n * 16;
  o[0] = depth; o[1] = m2x; o[2] = m2y; o[3] = c00; o[4] = c11; o[5] = cxy;
  o[6] = opacity[n];
  o[7] = rminx; o[8] = rminy; o[9] = rmaxx; o[10] = rmaxy;
  o[11] = col[0]; o[12] = col[1]; o[13] = col[2];
  o[14] = 0.f; o[15] = 0.f;
}

// ------------------------------------------------------------------
// Kernel 2: single-block bitonic argsort by depth (ascending)
// ------------------------------------------------------------------
__global__ __launch_bounds__(1024) void gs_sort(const float* __restrict__ pre,
                                                int* __restrict__ sidx) {
  __shared__ float key[NG];
  __shared__ int   idx[NG];
  for (int i = threadIdx.x; i < NG; i += 1024) { key[i] = pre[(size_t)i*16]; idx[i] = i; }
  __syncthreads();
  for (int k = 2; k <= NG; k <<= 1) {
    for (int j = k >> 1; j > 0; j >>= 1) {
      for (int i = threadIdx.x; i < NG; i += 1024) {
        const int l = i ^ j;
        if (l > i) {
          const bool asc = ((i & k) == 0);
          const float ki = key[i], kl = key[l];
          if ((ki > kl) == asc) {
            key[i] = kl; key[l] = ki;
            const int t = idx[i]; idx[i] = idx[l]; idx[l] = t;
          }
        }
      }
      __syncthreads();
    }
  }
  for (int i = threadIdx.x; i < NG; i += 1024) sidx[i] = idx[i];
}

// ------------------------------------------------------------------
// Kernel 3: gather sorted structs + build WMMA B panels
// sg[g*12]   = {m2x,m2y,c00,c11,cxy,op,rminx,rminy,rmaxx,rmaxy,0,0}
// bpanel[(g>>2)*64 + n*4 + (g&3)] : n=0..2 rgb, n=3 depth, n=4 one, n=5..15 zero
// ------------------------------------------------------------------
__global__ __launch_bounds__(256) void gs_gather(const float* __restrict__ pre,
                                                 const int* __restrict__ sidx,
                                                 float* __restrict__ sg,
                                                 float* __restrict__ bpanel) {
  const int g = blockIdx.x * 256 + threadIdx.x;
  if (g >= NG) return;
  const float* p = pre + (size_t)sidx[g] * 16;
  float* o = sg + (size_t)g * 12;
  o[0] = p[1];  o[1] = p[2];  o[2] = p[3];  o[3] = p[4];  o[4] = p[5];  o[5] = p[6];
  o[6] = p[7];  o[7] = p[8];  o[8] = p[9];  o[9] = p[10]; o[10] = 0.f;  o[11] = 0.f;
  float* bp = bpanel + (size_t)(g >> 2) * 64 + (g & 3);
  bp[0*4] = p[11]; bp[1*4] = p[12]; bp[2*4] = p[13]; bp[3*4] = p[0]; bp[4*4] = 1.0f;
#pragma unroll
  for (int nn = 5; nn < 16; ++nn) bp[nn*4] = 0.f;
}

// ------------------------------------------------------------------
// Kernel 4: tile render. 64 blocks (tiles), 256 threads = 8 waves,
// each wave owns 32 pixels. Chunks of K=4 sorted gaussians feed
// v_wmma_f32_16x16x4_f32: D[16px][16ch] += A[16px][4] * B[4][16ch].
// Both the gaussian structs and the B panels for a 256-gaussian
// superchunk are staged in LDS, so the inner loop is LDS + WMMA only.
// ------------------------------------------------------------------
__global__ __launch_bounds__(256) void gs_render(const float* __restrict__ sg,
                                                 const float* __restrict__ bpanel,
                                                 float* __restrict__ out) {
  const int tile = blockIdx.x;              // 0..63
  const int w0 = (tile & 7) * TIL;
  const int h0 = (tile >> 3) * TIL;
  const int tid  = threadIdx.x;
  const int wave = tid >> 5;                // 0..7
  const int lane = tid & 31;                // wave32
  const int half = lane >> 4;
  const int l15  = lane & 15;

  __shared__ __align__(16) float sgs[256 * 12];    // 12 KB gaussian superchunk
  __shared__ __align__(16) float bstage[64 * 64];  // 16 KB B panels (64 chunks)
  __shared__ __align__(16) float wbuf[8 * 128];    // 4 KB per-wave wgt scratch

  const int pix = wave * 32 + lane;               // pixel id in tile (row-major 16x16)
  const float px = (float)(w0 + (pix & 15));
  const float py = (float)(h0 + (pix >> 4));
  const float tminx = (float)w0, tminy = (float)h0;
  const float tmaxx = (float)(w0 + TIL - 1), tmaxy = (float)(h0 + TIL - 1);

  float T = 1.0f;
  v8f acc0 = {0.f,0.f,0.f,0.f,0.f,0.f,0.f,0.f};   // pixels wave*32 + 0..15
  v8f acc1 = {0.f,0.f,0.f,0.f,0.f,0.f,0.f,0.f};   // pixels wave*32 + 16..31

  float* const wbW = &wbuf[wave * 128];

  for (int base = 0; base < NG; base += 256) {
    __syncthreads();                               // protect previous superchunk
    {   // cooperative stage: gaussian structs (768 float4) + B panels (1024 float4)
      const v4f* srcA = (const v4f*)(sg + (size_t)base * 12);
      v4f* dstA = (v4f*)sgs;
      for (int i = tid; i < 768; i += 256) dstA[i] = srcA[i];
      const v4f* srcB = (const v4f*)(bpanel + (size_t)(base >> 2) * 64);
      v4f* dstB = (v4f*)bstage;
      for (int i = tid; i < 1024; i += 256) dstB[i] = srcB[i];
    }
    // prefetch next superchunk into near cache (global_prefetch_b8)
    if (base + 256 < NG) {
      const char* nsg = (const char*)(sg + (size_t)(base + 256) * 12);      // 12 KB
      const char* nbp = (const char*)(bpanel + (size_t)((base + 256) >> 2) * 64); // 16 KB
      if (tid < 96)       __builtin_prefetch(nsg + tid * 128, 0, 0);
      else if (tid < 224) __builtin_prefetch(nbp + (tid - 96) * 128, 0, 0);
    }
    __syncthreads();

    for (int ck = 0; ck < 64; ++ck) {              // 4-gaussian chunks
      float w4[4];
#pragma unroll
      for (int k = 0; k < 4; ++k) {
        const float* g = &sgs[(ck * 4 + k) * 12];
        const float tlx = fmaxf(g[6], tminx), tly = fmaxf(g[7], tminy);
        const float brx = fminf(g[8], tmaxx), bry = fminf(g[9], tmaxy);
        const float mask = ((brx > tlx) && (bry > tly)) ? 1.0f : 0.0f;
        const float ddx = px - g[0], ddy = py - g[1];
        const float power = -0.5f * (ddx*ddx*g[2] + ddy*ddy*g[3] + ddx*ddy*g[4]);
        const float alpha = fminf(expf(power) * g[5], 0.99f) * mask;
        w4[k] = alpha * T;
        T *= (1.0f - alpha);
      }
      // wave-private LDS reshape into WMMA A layout (lane = pixel-in-wave)
      *(v4f*)&wbW[lane * 4] = *(const v4f*)w4;
      __builtin_amdgcn_wave_barrier();
      // A layout (32-bit 16x4): lanes 0-15 -> K={0,1}, lanes 16-31 -> K={2,3}
      const v2f a0 = *(const v2f*)&wbW[l15 * 4 + half * 2];        // pixels 0..15
      const v2f a1 = *(const v2f*)&wbW[64 + l15 * 4 + half * 2];   // pixels 16..31
      __builtin_amdgcn_wave_barrier();
      // B layout (4x16): lane n=l15, K pair selected by half; panel stored [n][k]
      const v2f bmat = *(const v2f*)&bstage[ck * 64 + l15 * 4 + half * 2];
      acc0 = __builtin_amdgcn_wmma_f32_16x16x4_f32(false, a0, false, bmat,
                                                   (short)0, acc0, false, false);
      acc1 = __builtin_amdgcn_wmma_f32_16x16x4_f32(false, a1, false, bmat,
                                                   (short)0, acc1, false, false);
    }
  }

  // Epilogue. C layout: VGPR j holds (M = j + 8*half, N = l15).
  // N: 0..2 = rgb, 3 = depth, 4 = acc. White background: col += (1-acc).
#pragma unroll
  for (int gi = 0; gi < 2; ++gi) {
    const v8f c = gi ? acc1 : acc0;
    const int pixGroup = wave * 32 + gi * 16;
#pragma unroll
    for (int j = 0; j < 8; ++j) {
      const float v = c[j];
      const float accv = __shfl(v, (lane & 16) + 4, 32);   // acc channel, same M
      const int m = j + 8 * half;
      const int p = pixGroup + m;
      const int xg = w0 + (p & 15);
      const int yg = h0 + (p >> 4);
      if (l15 < 3) {
        out[(yg * IMW + xg) * 3 + l15] = v + (1.0f - accv);
      } else if (l15 == 3) {
        out[IMW * IMH * 3 + yg * IMW + xg] = v;
      } else if (l15 == 4) {
        out[IMW * IMH * 4 + yg * IMW + xg] = v;
      }
    }
  }
}

// ------------------------------------------------------------------
extern "C" void kernel_launch(void* const* d_in, const int* in_sizes, int n_in,
                              void* d_out, int out_size, void* d_ws, size_t ws_size,
                              hipStream_t stream) {
  const float* means3D   = (const float*)d_in[0];
  const float* opacity   = (const float*)d_in[1];
  const float* scales    = (const float*)d_in[2];
  const float* rotations = (const float*)d_in[3];
  const float* shs       = (const float*)d_in[4];
  const float* viewm     = (const float*)d_in[5];
  const float* projm     = (const float*)d_in[6];
  const float* camc      = (const float*)d_in[7];

  char* ws = (char*)d_ws;
  float* pre    = (float*)(ws + PRE_OFF);
  int*   sidx   = (int*)  (ws + SIDX_OFF);
  float* sgbuf  = (float*)(ws + SG_OFF);
  float* bpanel = (float*)(ws + BP_OFF);

  gs_preprocess<<<NG / 256, 256, 0, stream>>>(means3D, opacity, scales, rotations,
                                              shs, viewm, projm, camc, pre);
  gs_sort<<<1, 1024, 0, stream>>>(pre, sidx);
  gs_gather<<<NG / 256, 256, 0, stream>>>(pre, sidx, sgbuf, bpanel);
  gs_render<<<64, 256, 0, stream>>>(sgbuf, bpanel, (float*)d_out);
}